// CapsuleFC_9783935500865
// MI455X (gfx1250) — compile-verified
//
#include <hip/hip_runtime.h>
#include <math.h>

typedef float v2f __attribute__((ext_vector_type(2)));
typedef float v8f __attribute__((ext_vector_type(8)));

#define B_SZ   16
#define N_INC  1024
#define D_INC  16
#define N_OUTC 512
#define SQ     4
#define NCH    16
#define CHUNK  (N_INC / NCH)   // 64 n per chunk

// ws layout (floats):
//   [0, 16384)                : per-(b,n) row max of agreement
//   [16384, 32768)            : per-(b,n) 1/sum(exp) of agreement
//   [32768, 32768 + 16*131072): partial nv, [nc][b][m][ad]

// ---------------------------------------------------------------------------
// Kernel A: one block per n. 4 waves, wave rb handles b in [4rb, 4rb+4).
// Rows of the WMMA A-tile = (bsub, a); columns = (d fixed, 16 m's).
// After 4 WMMAs (d = 0..3) each lane holds all 16 (a,d) values for 2 b's at
// its column m -> lane-local softmax + dot with V16 -> agreement.
// ---------------------------------------------------------------------------
__global__ void __launch_bounds__(128)
caps_agreement_kernel(const float* __restrict__ x,
                      const float* __restrict__ w,
                      const float* __restrict__ V,
                      float* __restrict__ agr) {
  __shared__ float Vlds[N_OUTC * 17];          // pad 17 to kill bank conflicts
  const int n   = blockIdx.x;
  const int tid = threadIdx.x;
  for (int i = tid; i < N_OUTC * D_INC; i += 128)
    Vlds[(i >> 4) * 17 + (i & 15)] = V[i];
  __syncthreads();

  const int wave = tid >> 5;
  const int lane = tid & 31;
  const int h    = lane >> 4;                  // lane half (K pair / row half)
  const int ll   = lane & 15;
  const int rb   = wave;                       // b-group
  const int bsub = ll >> 2;
  const int a    = ll & 3;
  const int brow = rb * 4 + bsub;
  const int x0   = h * 2;

  // A fragment: A[row=ll, K=x0+v] = in[brow, n, a, x0+v]  (2 consecutive floats)
  const float* ap = x + (((size_t)brow * N_INC + n) * D_INC + a * SQ + x0);
  v2f afrag;
  afrag.x = ap[0];
  afrag.y = ap[1];

  for (int mt = 0; mt < 32; ++mt) {
    const int mcol = mt * 16 + ll;
    v8f dd[4];
#pragma unroll
    for (int d = 0; d < 4; ++d) {
      // B fragment: B[K=x0+v, N=ll] = w[n, x0+v, d, mcol]
      const float* bp = w + ((((size_t)n * SQ + x0) * SQ + d) * N_OUTC + mcol);
      v2f bfrag;
      bfrag.x = bp[0];
      bfrag.y = bp[SQ * N_OUTC];               // next x -> stride 4*512 floats
      v8f c = {};
      dd[d] = __builtin_amdgcn_wmma_f32_16x16x4_f32(
          false, afrag, false, bfrag, (short)0, c, false, false);
    }
#pragma unroll
    for (int s = 0; s < 2; ++s) {              // two b values per lane
      float v[16];
#pragma unroll
      for (int d = 0; d < 4; ++d)
#pragma unroll
        for (int aa = 0; aa < 4; ++aa)
          v[aa * 4 + d] = dd[d][s * 4 + aa];   // row M = h*8 + s*4 + aa
      float mx = v[0];
#pragma unroll
      for (int i = 1; i < 16; ++i) mx = fmaxf(mx, v[i]);
      float sum = 0.f, num = 0.f;
#pragma unroll
      for (int i = 0; i < 16; ++i) {
        float e = __expf(v[i] - mx);
        sum += e;
        num += e * Vlds[mcol * 17 + i];
      }
      const int bb = rb * 4 + h * 2 + s;
      agr[((size_t)bb * N_INC + n) * N_OUTC + mcol] = num / sum;
    }
  }
}

// ---------------------------------------------------------------------------
// Kernel B: softmax stats over m=512 per (b,n) row: max and 1/sum(exp).
// One wave per row, 8 waves per block.
// ---------------------------------------------------------------------------
__global__ void __launch_bounds__(256)
caps_qkstats_kernel(const float* __restrict__ agr, float* __restrict__ ws) {
  const int wave = threadIdx.x >> 5;
  const int lane = threadIdx.x & 31;
  const int row  = blockIdx.x * 8 + wave;      // (b*1024 + n), 0..16383
  const float* p = agr + (size_t)row * N_OUTC;
  float vals[16];
  float mx = -3.4e38f;
#pragma unroll
  for (int k = 0; k < 16; ++k) {
    vals[k] = p[k * 32 + lane];
    mx = fmaxf(mx, vals[k]);
  }
#pragma unroll
  for (int off = 16; off >= 1; off >>= 1) mx = fmaxf(mx, __shfl_xor(mx, off, 32));
  float sum = 0.f;
#pragma unroll
  for (int k = 0; k < 16; ++k) sum += __expf(vals[k] - mx);
#pragma unroll
  for (int off = 16; off >= 1; off >>= 1) sum += __shfl_xor(sum, off, 32);
  if (lane == 0) {
    ws[row]         = mx;
    ws[16384 + row] = 1.0f / sum;
  }
}

// ---------------------------------------------------------------------------
// Kernel C: block per (m-tile, n-chunk). Recompute Vs via WMMA (w is L2
// resident), weight by qk = exp(agr - max) * rinv, accumulate partial
// nv[b, m, ad] in registers over 64 n's; write to ws partials.
// ---------------------------------------------------------------------------
__global__ void __launch_bounds__(128)
caps_accum_kernel(const float* __restrict__ x,
                  const float* __restrict__ w,
                  const float* __restrict__ agr,
                  const float* __restrict__ ws,
                  float* __restrict__ part) {
  const int mt   = blockIdx.x;                 // 0..31
  const int nc   = blockIdx.y;                 // 0..NCH-1
  const int tid  = threadIdx.x;
  const int wave = tid >> 5;
  const int lane = tid & 31;
  const int h    = lane >> 4;
  const int ll   = lane & 15;
  const int rb   = wave;
  const int bsub = ll >> 2;
  const int a    = ll & 3;
  const int brow = rb * 4 + bsub;
  const int x0   = h * 2;
  const int mcol = mt * 16 + ll;
  const int b0   = rb * 4 + h * 2;             // first of this lane's two b's

  float acc[2][16];
#pragma unroll
  for (int s = 0; s < 2; ++s)
#pragma unroll
    for (int i = 0; i < 16; ++i) acc[s][i] = 0.f;

  for (int ni = 0; ni < CHUNK; ++ni) {
    const int n = nc * CHUNK + ni;
    const float* ap = x + (((size_t)brow * N_INC + n) * D_INC + a * SQ + x0);
    v2f afrag;
    afrag.x = ap[0];
    afrag.y = ap[1];

    v8f dd[4];
#pragma unroll
    for (int d = 0; d < 4; ++d) {
      const float* bp = w + ((((size_t)n * SQ + x0) * SQ + d) * N_OUTC + mcol);
      v2f bfrag;
      bfrag.x = bp[0];
      bfrag.y = bp[SQ * N_OUTC];
      v8f c = {};
      dd[d] = __builtin_amdgcn_wmma_f32_16x16x4_f32(
          false, afrag, false, bfrag, (short)0, c, false, false);
    }

    float qk[2];
#pragma unroll
    for (int s = 0; s < 2; ++s) {
      const int row = (b0 + s) * N_INC + n;
      const float ag = agr[(size_t)row * N_OUTC + mcol];
      qk[s] = __expf(ag - ws[row]) * ws[16384 + row];
    }

#pragma unroll
    for (int s = 0; s < 2; ++s) {
      float v[16];
#pragma unroll
      for (int d = 0; d < 4; ++d)
#pragma unroll
        for (int aa = 0; aa < 4; ++aa)
          v[aa * 4 + d] = dd[d][s * 4 + aa];
      float mx = v[0];
#pragma unroll
      for (int i = 1; i < 16; ++i) mx = fmaxf(mx, v[i]);
      float e[16];
      float sum = 0.f;
#pragma unroll
      for (int i = 0; i < 16; ++i) {
        e[i] = __expf(v[i] - mx);
        sum += e[i];
      }
      const float scale = qk[s] / sum;         // qk * Vs normalization
#pragma unroll
      for (int i = 0; i < 16; ++i) acc[s][i] += e[i] * scale;
    }
  }

#pragma unroll
  for (int s = 0; s < 2; ++s) {
    float* pp = part + ((((size_t)nc * 16 + (b0 + s)) * N_OUTC + mcol) << 4);
#pragma unroll
    for (int i = 0; i < 16; ++i) pp[i] = acc[s][i];
  }
}

// ---------------------------------------------------------------------------
// Kernel D: reduce the NCH partials and apply LayerNorm over the 16-dim pose.
// One thread per (b, m).
// ---------------------------------------------------------------------------
__global__ void __launch_bounds__(256)
caps_ln_kernel(const float* __restrict__ part,
               const float* __restrict__ gamma,
               const float* __restrict__ beta,
               float* __restrict__ nv) {
  const int t = blockIdx.x * 256 + threadIdx.x;   // 0..8191 = b*512 + m
  float s[16];
#pragma unroll
  for (int i = 0; i < 16; ++i) s[i] = 0.f;
  for (int nc = 0; nc < NCH; ++nc) {
    const float* p = part + (((size_t)nc * (B_SZ * N_OUTC) + t) << 4);
#pragma unroll
    for (int i = 0; i < 16; ++i) s[i] += p[i];
  }
  float mu = 0.f;
#pragma unroll
  for (int i = 0; i < 16; ++i) mu += s[i];
  mu *= (1.0f / 16.0f);
  float var = 0.f;
#pragma unroll
  for (int i = 0; i < 16; ++i) {
    const float d = s[i] - mu;
    var += d * d;
  }
  var *= (1.0f / 16.0f);
  const float rs = rsqrtf(var + 1e-5f);
#pragma unroll
  for (int i = 0; i < 16; ++i)
    nv[(size_t)t * 16 + i] = (s[i] - mu) * rs * gamma[i] + beta[i];
}

// ---------------------------------------------------------------------------
extern "C" void kernel_launch(void* const* d_in, const int* in_sizes, int n_in,
                              void* d_out, int out_size, void* d_ws, size_t ws_size,
                              hipStream_t stream) {
  (void)in_sizes; (void)n_in; (void)out_size; (void)ws_size;
  const float* x     = (const float*)d_in[0];   // [16,1024,16]
  // d_in[1] = num_iter (unused by reference math)
  const float* w     = (const float*)d_in[2];   // [1024,4,4,512]
  const float* V     = (const float*)d_in[3];   // [512,16]
  const float* gamma = (const float*)d_in[4];   // [16]
  const float* beta  = (const float*)d_in[5];   // [16]

  float* nv  = (float*)d_out;                   // [16,512,16]   (output 0)
  float* agr = nv + B_SZ * N_OUTC * D_INC;      // [16,1024,512] (output 1)

  float* ws   = (float*)d_ws;                   // stats
  float* part = ws + 32768;                     // partial nv

  caps_agreement_kernel<<<N_INC, 128, 0, stream>>>(x, w, V, agr);
  caps_qkstats_kernel<<<(B_SZ * N_INC) / 8, 256, 0, stream>>>(agr, ws);
  caps_accum_kernel<<<dim3(32, NCH), 128, 0, stream>>>(x, w, agr, ws, part);
  caps_ln_kernel<<<(B_SZ * N_OUTC) / 256, 256, 0, stream>>>(part, gamma, beta, nv);
}